// BatchAllTripletLoss_58454504898917
// MI455X (gfx1250) — compile-verified
//
#include <hip/hip_runtime.h>

typedef float v2f __attribute__((ext_vector_type(2)));
typedef float v8f __attribute__((ext_vector_type(8)));

#define NROW 512
#define HALF_N 256
#define DIM 256
#define EPS_DIST 1e-7f
#define EPS_REL 1e-5f

// ---------------------------------------------------------------------------
// Kernel 1: sq[r] = sum_k batch[r][k]^2   (batch = concat(h1, h2))
// ---------------------------------------------------------------------------
__global__ void __launch_bounds__(256)
row_norms_kernel(const float* __restrict__ h1, const float* __restrict__ h2,
                 float* __restrict__ sq) {
    int r = blockIdx.x * blockDim.x + threadIdx.x;
    if (r >= NROW) return;
    const float* src = (r < HALF_N) ? (h1 + r * DIM) : (h2 + (r - HALF_N) * DIM);
    const float4* v = (const float4*)src;
    float acc = 0.0f;
#pragma unroll 8
    for (int k = 0; k < DIM / 4; ++k) {
        float4 x = v[k];
        acc += x.x * x.x + x.y * x.y + x.z * x.z + x.w * x.w;
    }
    sq[r] = acc;
}

// ---------------------------------------------------------------------------
// Kernel 2: dists[i][j] = clamp(sq[i] + sq[j] - 2 * <row_i, row_j>, 1e-7)
// One 16x16 output tile per wave, K-loop via V_WMMA_F32_16X16X4_F32.
// A fragment (16x4, MxK):  lane l (0..15), half h: holds A[M=l][K=2h+{0,1}]
// B fragment (4x16, KxN):  lane l, half h:          holds B[K=2h+{0,1}][N=l]
// Both map to an 8B load at srcRow(base+l) + k0 + 2h.
// ---------------------------------------------------------------------------
__global__ void __launch_bounds__(256)
gram_dist_kernel(const float* __restrict__ h1, const float* __restrict__ h2,
                 const float* __restrict__ sq, float* __restrict__ dists) {
    const int wave = threadIdx.x >> 5;
    const int lane = threadIdx.x & 31;
    const int tile = blockIdx.x * 8 + wave;   // 0..1023
    const int ti = tile >> 5;                 // 0..31  -> i0 = 16*ti
    const int tj = tile & 31;                 //        -> j0 = 16*tj
    const int i0 = ti * 16, j0 = tj * 16;
    const int l = lane & 15;
    const int h = lane >> 4;

    // 16-row tile never straddles the h1/h2 boundary (i0 multiple of 16).
    const float* abase = (i0 < HALF_N) ? (h1 + i0 * DIM) : (h2 + (i0 - HALF_N) * DIM);
    const float* bbase = (j0 < HALF_N) ? (h1 + j0 * DIM) : (h2 + (j0 - HALF_N) * DIM);
    const float* aptr = abase + l * DIM + 2 * h;
    const float* bptr = bbase + l * DIM + 2 * h;

    v8f c = {};
#pragma unroll 4
    for (int k0 = 0; k0 < DIM; k0 += 4) {
        v2f a = *(const v2f*)(aptr + k0);
        v2f b = *(const v2f*)(bptr + k0);
        // 8 args: (neg_a, A, neg_b, B, c_mod, C, reuse_a, reuse_b)
        c = __builtin_amdgcn_wmma_f32_16x16x4_f32(
                false, a, false, b, (short)0, c, false, false);
    }

    const float sqj = sq[j0 + l];
#pragma unroll
    for (int r = 0; r < 8; ++r) {
        const int row = i0 + r + 8 * h;       // C/D layout: VGPR r, half h -> M = r + 8h
        float d = sq[row] + sqj - 2.0f * c[r];
        d = fmaxf(d, 0.0f);                   // NaN->finite + clamp(min=0)
        d = fmaxf(d, EPS_DIST);               // clamp(min=1e-7)
        dists[row * NROW + (j0 + l)] = d;
    }
}

// ---------------------------------------------------------------------------
// Kernel 3: only triplet entries with mask==1 are (i, j, j^256).
// t = relu(dists[i][j] - dists[i][j^256] + 1)
// Single 1024-thread block; LDS tree reduction; thread 0 writes 5 outputs.
// ---------------------------------------------------------------------------
__global__ void __launch_bounds__(1024)
reduce_final_kernel(const float* __restrict__ dists, const float* __restrict__ sq,
                    float* __restrict__ out) {
    __shared__ float s_sum[1024];
    __shared__ float s_crel[1024];
    __shared__ float s_cgood[1024];
    __shared__ float s_sq[1024];

    const int t = threadIdx.x;
    float sumRel = 0.0f, cntRel = 0.0f, cntGood = 0.0f;

    for (int idx = t; idx < NROW * NROW; idx += 1024) {
        const int i = idx >> 9;
        const int j = idx & (NROW - 1);
        const float d1 = dists[i * NROW + j];
        const float d2 = dists[i * NROW + (j ^ HALF_N)];
        const float tt = fmaxf(d1 - d2 + 1.0f, 0.0f);
        if (tt > EPS_REL) { sumRel += tt; cntRel += 1.0f; }
        if (tt < EPS_REL) { cntGood += 1.0f; }
    }

    s_sum[t]   = sumRel;
    s_crel[t]  = cntRel;
    s_cgood[t] = cntGood;
    s_sq[t]    = (t < NROW) ? sq[t] : 0.0f;
    __syncthreads();

    for (int s = 512; s > 0; s >>= 1) {
        if (t < s) {
            s_sum[t]   += s_sum[t + s];
            s_crel[t]  += s_crel[t + s];
            s_cgood[t] += s_cgood[t + s];
            s_sq[t]    += s_sq[t + s];
        }
        __syncthreads();
    }

    if (t == 0) {
        const float mns = s_sq[0] / (float)NROW;                // mean(sq)
        const float meanRel = s_sum[0] / s_crel[0];
        const float totalCube = 134217728.0f;                   // 512^3
        const float maskedCnt = 262144.0f;                      // 512^2
        out[0] = meanRel + 1e-4f * mns;                         // loss
        out[1] = 0.0f;                                          // mean(differences) == 0
        out[2] = (totalCube - maskedCnt) + s_cgood[0];          // good
        out[3] = maskedCnt - s_cgood[0];                        // bad
        out[4] = sqrtf(mns);                                    // sqrt(mean_norm_squared)
    }
}

// ---------------------------------------------------------------------------
extern "C" void kernel_launch(void* const* d_in, const int* in_sizes, int n_in,
                              void* d_out, int out_size, void* d_ws, size_t ws_size,
                              hipStream_t stream) {
    const float* h1 = (const float*)d_in[0];
    const float* h2 = (const float*)d_in[1];
    // d_in[2] (h3) is unused by the reference.
    float* ws    = (float*)d_ws;
    float* dists = ws;                       // 512*512 floats = 1 MB
    float* sq    = ws + NROW * NROW;         // 512 floats
    float* out   = (float*)d_out;            // 5 floats

    row_norms_kernel<<<2, 256, 0, stream>>>(h1, h2, sq);
    gram_dist_kernel<<<128, 256, 0, stream>>>(h1, h2, sq, dists);
    reduce_final_kernel<<<1, 1024, 0, stream>>>(dists, sq, out);
}